// simple_GRU_cell_90623809946399
// MI455X (gfx1250) — compile-verified
//
#include <hip/hip_runtime.h>
#include <stdint.h>

// Problem constants (match reference).
#define B_SZ 2048
#define E_SZ 512
#define H_SZ 512
#define V_SZ 50257
#define LN_EPS 1e-5f

typedef __attribute__((ext_vector_type(2))) float v2f;
typedef __attribute__((ext_vector_type(8))) float v8f;

#define BM 64
#define BN 256
#define KC 32
#define KST 36  // padded row stride in floats: 144B rows -> 16B-aligned, skewed banks

#if defined(__gfx1250__) && __has_builtin(__builtin_amdgcn_global_load_async_to_lds_b128) && __has_builtin(__builtin_amdgcn_s_wait_asynccnt)
#define USE_ASYNC 1
#else
#define USE_ASYNC 0
#endif

// Builtin prototype uses GCC-vector int4 pointers with explicit address spaces:
//   void __builtin_amdgcn_global_load_async_to_lds_b128(v4i AS1*, v4i AS3*, Ii, Ii)
typedef int v4i __attribute__((vector_size(16)));
typedef __attribute__((address_space(1))) v4i glb_v4i;
typedef __attribute__((address_space(3))) v4i lds_v4i;

enum { A_PLAIN = 0, A_CONCAT = 1, A_CAND = 2 };

// C[M,N] = Aloader[M,K] @ W[N,K]^T + bias[N]
// Block tile 64x256, 8 wave32s; each wave owns 32x64 (8 WMMA tiles, 64 acc VGPRs).
// Double-buffered LDS; W tile streamed with async global->LDS DMA (ASYNCcnt),
// A tile via registers (needed for concat / h_prev*r transforms).
template <int MODE>
__global__ __launch_bounds__(256)
void gemm_wmma_f32(const float* __restrict__ Ah,   // h_prev (CONCAT/CAND) or h_t (PLAIN)
                   const float* __restrict__ Ax,   // x (CONCAT/CAND)
                   const float* __restrict__ Ar,   // r (CAND)
                   const float* __restrict__ W,    // [N,K] row-major
                   const float* __restrict__ bias, // [N]
                   float* __restrict__ C,          // [M,N]
                   int N, int K)
{
  __shared__ float As[2][BM][KST];   // [buf][m][k]
  __shared__ float Bs[2][BN][KST];   // [buf][n][k]  (k contiguous -> b64 frag loads)

  const int tid   = threadIdx.x;
  const int lane  = tid & 31;
  const int wv    = tid >> 5;
  const int waveM = wv >> 2;    // 0..1 -> 32-row group
  const int waveN = wv & 3;     // 0..3 -> 64-col group
  const int hf    = lane >> 4;  // lane half: A/B upper half holds K+2
  const int l16   = lane & 15;

  const int mBlock = blockIdx.y * BM;
  const int nBlock = blockIdx.x * BN;

  // Tile-load mapping: 8 threads cover KC via float4, 32 rows per pass.
  const int kLane = tid & 7;
  const int rIdx  = tid >> 3;   // 0..31
  const int kcol  = kLane * 4;

  const v8f vz = {0.f, 0.f, 0.f, 0.f, 0.f, 0.f, 0.f, 0.f};
  v8f acc[2][4];
#pragma unroll
  for (int i = 0; i < 2; ++i)
#pragma unroll
    for (int t = 0; t < 4; ++t) acc[i][t] = vz;

  float4 aReg[2];
#if !USE_ASYNC
  float4 bReg[8];
#endif

  auto load_a = [&](int k0) {
#pragma unroll
    for (int p = 0; p < 2; ++p) {
      const int gm = mBlock + rIdx + p * 32;
      const int gk = k0 + kcol;                 // mult of 4; never straddles 512
      float4 av;
      if (MODE == A_PLAIN) {
        av = *(const float4*)(Ah + (size_t)gm * K + gk);
      } else if (gk < H_SZ) {
        av = *(const float4*)(Ah + (size_t)gm * H_SZ + gk);
        if (MODE == A_CAND) {
          const float4 rv = *(const float4*)(Ar + (size_t)gm * H_SZ + gk);
          av.x *= rv.x; av.y *= rv.y; av.z *= rv.z; av.w *= rv.w;
        }
      } else {
        av = *(const float4*)(Ax + (size_t)gm * E_SZ + (gk - H_SZ));
      }
      aReg[p] = av;
    }
  };
  auto store_a = [&](int buf) {
#pragma unroll
    for (int p = 0; p < 2; ++p)
      *(float4*)&As[buf][rIdx + p * 32][kcol] = aReg[p];
  };
  // Weight tile: per-lane async DMA straight into LDS (or reg-staged fallback).
  // Out-of-range vocab rows are clamped (harmless: their columns are never stored).
  auto issue_b = [&](int buf, int k0) {
#pragma unroll
    for (int p = 0; p < 8; ++p) {
      const int n = rIdx + p * 32;
      int gn = nBlock + n;
      if (gn >= N) gn = N - 1;
      const float* src = W + (size_t)gn * K + k0 + kcol;
#if USE_ASYNC
      __builtin_amdgcn_global_load_async_to_lds_b128(
          (glb_v4i*)(uintptr_t)src,
          (lds_v4i*)(uintptr_t)&Bs[buf][n][kcol], 0, 0);
#else
      bReg[p] = *(const float4*)src;
#endif
    }
    (void)buf;
  };
  auto commit_b = [&](int buf) {
#if !USE_ASYNC
#pragma unroll
    for (int p = 0; p < 8; ++p)
      *(float4*)&Bs[buf][rIdx + p * 32][kcol] = bReg[p];
#endif
    (void)buf;
  };
  auto wait_async = [&]() {
#if USE_ASYNC
    __builtin_amdgcn_s_wait_asynccnt(0);
#endif
  };

  // ---- Prologue: stage chunk 0.
  issue_b(0, 0);
  load_a(0);
  store_a(0);
  commit_b(0);
  wait_async();
  __syncthreads();

  const int ktiles = K / KC;
  for (int kt = 0; kt < ktiles; ++kt) {
    const int cur  = kt & 1;
    const bool more = (kt + 1) < ktiles;
    if (more) {                       // overlap next-chunk fetch with compute
      issue_b(cur ^ 1, (kt + 1) * KC);
      load_a((kt + 1) * KC);
    }

    const int mW = waveM * 32 + l16;
    const int nW = waveN * 64 + l16;
#pragma unroll
    for (int kk = 0; kk < KC; kk += 4) {
      const int ka = kk + 2 * hf;     // lanes 0-15: K,K+1 ; lanes 16-31: K+2,K+3
      v2f a0, a1;
      a0.x = As[cur][mW][ka];       a0.y = As[cur][mW][ka + 1];
      a1.x = As[cur][mW + 16][ka];  a1.y = As[cur][mW + 16][ka + 1];
#pragma unroll
      for (int t = 0; t < 4; ++t) {
        v2f b;
        b.x = Bs[cur][nW + t * 16][ka];
        b.y = Bs[cur][nW + t * 16][ka + 1];
        acc[0][t] = __builtin_amdgcn_wmma_f32_16x16x4_f32(false, a0, false, b, (short)0, acc[0][t], false, false);
        acc[1][t] = __builtin_amdgcn_wmma_f32_16x16x4_f32(false, a1, false, b, (short)0, acc[1][t], false, false);
      }
    }

    if (more) {
      store_a(cur ^ 1);
      commit_b(cur ^ 1);
      wait_async();
      __syncthreads();                // single barrier per K-chunk
    }
  }

  // ---- Epilogue: C VGPR j holds rows M=j (lanes 0-15) and M=j+8 (lanes 16-31).
#pragma unroll
  for (int t = 0; t < 4; ++t) {
    const int gn = nBlock + waveN * 64 + t * 16 + l16;
    if (gn < N) {
      const float bv = bias[gn];
#pragma unroll
      for (int i = 0; i < 2; ++i) {
#pragma unroll
        for (int j = 0; j < 8; ++j) {
          const int gm = mBlock + waveM * 32 + i * 16 + hf * 8 + j;
          C[(size_t)gm * N + gn] = acc[i][t][j] + bv;
        }
      }
    }
  }
}

// Row LayerNorm (over H=512) + sigmoid. One 256-thread block per row; 2 elems/thread.
__global__ __launch_bounds__(256)
void ln_sigmoid(const float* __restrict__ pre, const float* __restrict__ g,
                const float* __restrict__ be, float* __restrict__ out)
{
  __shared__ float s1[256];
  __shared__ float s2[256];
  const int row = blockIdx.x;
  const int tid = threadIdx.x;
  const float* p = pre + (size_t)row * H_SZ;
  const float v0 = p[tid];
  const float v1 = p[tid + 256];
  s1[tid] = v0 + v1;
  s2[tid] = v0 * v0 + v1 * v1;
  __syncthreads();
  for (int st = 128; st > 0; st >>= 1) {
    if (tid < st) { s1[tid] += s1[tid + st]; s2[tid] += s2[tid + st]; }
    __syncthreads();
  }
  const float mean = s1[0] * (1.0f / H_SZ);
  const float var  = s2[0] * (1.0f / H_SZ) - mean * mean;
  const float rstd = rsqrtf(var + LN_EPS);
  const float o0 = (v0 - mean) * rstd * g[tid] + be[tid];
  const float o1 = (v1 - mean) * rstd * g[tid + 256] + be[tid + 256];
  out[(size_t)row * H_SZ + tid]       = 1.0f / (1.0f + expf(-o0));
  out[(size_t)row * H_SZ + tid + 256] = 1.0f / (1.0f + expf(-o1));
}

// Row LayerNorm + tanh, then h_t = (1-z)*h_prev + z*h_new.
__global__ __launch_bounds__(256)
void ln_tanh_combine(const float* __restrict__ pre, const float* __restrict__ g,
                     const float* __restrict__ be, const float* __restrict__ zg,
                     const float* __restrict__ h_prev, float* __restrict__ h_out)
{
  __shared__ float s1[256];
  __shared__ float s2[256];
  const int row = blockIdx.x;
  const int tid = threadIdx.x;
  const size_t base = (size_t)row * H_SZ;
  const float v0 = pre[base + tid];
  const float v1 = pre[base + tid + 256];
  s1[tid] = v0 + v1;
  s2[tid] = v0 * v0 + v1 * v1;
  __syncthreads();
  for (int st = 128; st > 0; st >>= 1) {
    if (tid < st) { s1[tid] += s1[tid + st]; s2[tid] += s2[tid + st]; }
    __syncthreads();
  }
  const float mean = s1[0] * (1.0f / H_SZ);
  const float var  = s2[0] * (1.0f / H_SZ) - mean * mean;
  const float rstd = rsqrtf(var + LN_EPS);
  const float hn0 = tanhf((v0 - mean) * rstd * g[tid] + be[tid]);
  const float hn1 = tanhf((v1 - mean) * rstd * g[tid + 256] + be[tid + 256]);
  const float z0 = zg[base + tid];
  const float z1 = zg[base + tid + 256];
  h_out[base + tid]       = (1.0f - z0) * h_prev[base + tid]       + z0 * hn0;
  h_out[base + tid + 256] = (1.0f - z1) * h_prev[base + tid + 256] + z1 * hn1;
}

extern "C" void kernel_launch(void* const* d_in, const int* in_sizes, int n_in,
                              void* d_out, int out_size, void* d_ws, size_t ws_size,
                              hipStream_t stream)
{
  (void)in_sizes; (void)n_in; (void)out_size; (void)d_ws; (void)ws_size;

  const float* x      = (const float*)d_in[0];
  const float* h_prev = (const float*)d_in[1];
  const float* W_z  = (const float*)d_in[2];
  const float* b_z  = (const float*)d_in[3];
  const float* g_z  = (const float*)d_in[4];
  const float* be_z = (const float*)d_in[5];
  const float* W_r  = (const float*)d_in[6];
  const float* b_r  = (const float*)d_in[7];
  const float* g_r  = (const float*)d_in[8];
  const float* be_r = (const float*)d_in[9];
  const float* W_h  = (const float*)d_in[10];
  const float* b_h  = (const float*)d_in[11];
  const float* g_h  = (const float*)d_in[12];
  const float* be_h = (const float*)d_in[13];
  const float* W_o  = (const float*)d_in[14];
  const float* b_o  = (const float*)d_in[15];

  float* h_out = (float*)d_out;                    // [B,H]
  float* y_out = h_out + (size_t)B_SZ * H_SZ;      // [B,V]

  // Scratch lives in the (huge) y_t region; final GEMM overwrites it last.
  float* Zpre = y_out;
  float* Rpre = Zpre + (size_t)B_SZ * H_SZ;
  float* zg   = Rpre + (size_t)B_SZ * H_SZ;
  float* rg   = zg   + (size_t)B_SZ * H_SZ;
  float* Hpre = rg   + (size_t)B_SZ * H_SZ;

  const dim3 blk(256);
  const dim3 gGate(H_SZ / BN, B_SZ / BM);                  // (2, 32)

  gemm_wmma_f32<A_CONCAT><<<gGate, blk, 0, stream>>>(
      h_prev, x, nullptr, W_z, b_z, Zpre, H_SZ, H_SZ + E_SZ);
  gemm_wmma_f32<A_CONCAT><<<gGate, blk, 0, stream>>>(
      h_prev, x, nullptr, W_r, b_r, Rpre, H_SZ, H_SZ + E_SZ);

  ln_sigmoid<<<dim3(B_SZ), blk, 0, stream>>>(Zpre, g_z, be_z, zg);
  ln_sigmoid<<<dim3(B_SZ), blk, 0, stream>>>(Rpre, g_r, be_r, rg);

  gemm_wmma_f32<A_CAND><<<gGate, blk, 0, stream>>>(
      h_prev, x, rg, W_h, b_h, Hpre, H_SZ, H_SZ + E_SZ);

  ln_tanh_combine<<<dim3(B_SZ), blk, 0, stream>>>(Hpre, g_h, be_h, zg, h_prev, h_out);

  const dim3 gOut((V_SZ + BN - 1) / BN, B_SZ / BM);        // (197, 32)
  gemm_wmma_f32<A_PLAIN><<<gOut, blk, 0, stream>>>(
      h_out, nullptr, nullptr, W_o, b_o, y_out, V_SZ, H_SZ);
}